// SubLinear_66271345377403
// MI455X (gfx1250) — compile-verified
//
#include <hip/hip_runtime.h>

#define TPB 256
#define CHUNK 8192
#define MAX_OUT 64

typedef __attribute__((ext_vector_type(2))) float v2f;
typedef __attribute__((ext_vector_type(8))) float v8f;
typedef int v4i_vs __attribute__((vector_size(16)));   // matches builtin param

// ---------------------------------------------------------------------------
// Kernel 0: find segment start offsets (segments are contiguous, ascending).
// starts[s] = first feature index of segment s; starts[nOut] = F sentinel.
// ---------------------------------------------------------------------------
__global__ void seg_starts_kernel(const int* __restrict__ seg, int F,
                                  int* __restrict__ starts, int nOut) {
  const int stride = gridDim.x * blockDim.x;
  for (int i = blockIdx.x * blockDim.x + threadIdx.x; i < F; i += stride) {
    const int s = seg[i];
    if (i == 0) {
      starts[s] = 0;
    } else if (seg[i - 1] != s) {
      starts[s] = i;
    }
  }
  if (blockIdx.x == 0 && threadIdx.x == 0) starts[nOut] = F;
}

// ---------------------------------------------------------------------------
// Kernel 1: bandwidth-bound streaming segment-sum.
// Each block owns an 8192-feature chunk, stages w into LDS once (via
// GLOBAL_LOAD_ASYNC_TO_LDS_B128 when available), then each of the 8 waves
// streams 8 batch rows with coalesced float4 loads. Per-lane monotone run
// accumulation; boundary crossings flush via LDS float atomics.
// Writes partial[blk][row][0..nOut) to workspace (deterministic).
// ---------------------------------------------------------------------------
__global__ __launch_bounds__(TPB)
void seglinear_kernel(const float* __restrict__ x, const float* __restrict__ w,
                      const int* __restrict__ starts, float* __restrict__ partial,
                      int F, int Bn, int nOut) {
  __shared__ float w_lds[CHUNK];
  __shared__ float wacc[TPB / 32][MAX_OUT];
  __shared__ int sstarts[MAX_OUT + 1];

  const int tid  = threadIdx.x;
  const int lane = tid & 31;
  const int wave = tid >> 5;
  const int base = blockIdx.x * CHUNK;
  const int rem  = F - base;
  const int cend = (rem < CHUNK) ? rem : CHUNK;

  if (tid <= nOut) sstarts[tid] = starts[tid];
  for (int t = tid; t < (TPB / 32) * MAX_OUT; t += TPB)
    (&wacc[0][0])[t] = 0.f;

  // ---- stage the w chunk into LDS (reused across all batch rows) ----
  const int nv = cend >> 2;
#if __has_builtin(__builtin_amdgcn_global_load_async_to_lds_b128)
  for (int t = tid; t < nv; t += TPB) {
    const float* gsrc = w + base + (t << 2);
    float*       ldst = w_lds + (t << 2);
    // AS1 value == flat address; AS3 value == flat address low 32 bits (LDS
    // byte offset; aperture lives in the high half per the ISA).
    __builtin_amdgcn_global_load_async_to_lds_b128(
        (__attribute__((address_space(1))) v4i_vs*)(uintptr_t)gsrc,
        (__attribute__((address_space(3))) v4i_vs*)(uint32_t)(uintptr_t)ldst,
        0, 0);
  }
  for (int t = (nv << 2) + tid; t < cend; t += TPB) w_lds[t] = w[base + t];
#if __has_builtin(__builtin_amdgcn_s_wait_asynccnt)
  __builtin_amdgcn_s_wait_asynccnt(0);
#else
  asm volatile("s_wait_asynccnt 0" ::: "memory");
#endif
#else
  for (int t = tid; t < nv; t += TPB)
    reinterpret_cast<float4*>(w_lds)[t] =
        reinterpret_cast<const float4*>(w + base)[t];
  for (int t = (nv << 2) + tid; t < cend; t += TPB) w_lds[t] = w[base + t];
#endif
  __syncthreads();

  // initial segment for this lane's first element (binary search, done once)
  int cur0;
  {
    int gi0 = base + (lane << 2);
    if (gi0 >= F) gi0 = F - 1;
    int lo = 0, hi = nOut - 1;
    while (lo < hi) {
      const int mid = (lo + hi + 1) >> 1;
      if (sstarts[mid] <= gi0) lo = mid; else hi = mid - 1;
    }
    cur0 = lo;
  }

  const int nIter    = cend >> 7;                       // vec4 * 32 lanes
  const int rowTrips = (Bn + (TPB / 32) - 1) / (TPB / 32);

  for (int rr = 0; rr < rowTrips; ++rr) {
    const int row = rr * (TPB / 32) + wave;
    if (row < Bn) {
      const float* __restrict__ xrow = x + (size_t)row * F + base;
      int cur = cur0;
      int nb  = sstarts[cur + 1];
      float acc = 0.f;
      for (int it = 0; it < nIter; ++it) {
        const int li = (it << 7) + (lane << 2);
        __builtin_prefetch(xrow + li + 4096, 0, 0);      // global_prefetch_b8
        const float4 xv = *reinterpret_cast<const float4*>(xrow + li);
        const float4 wv = *reinterpret_cast<const float4*>(w_lds + li);
        if (base + li + 3 < nb) {                        // common: one segment
          acc = fmaf(xv.x, wv.x, acc);
          acc = fmaf(xv.y, wv.y, acc);
          acc = fmaf(xv.z, wv.z, acc);
          acc = fmaf(xv.w, wv.w, acc);
        } else {                                         // boundary crossing
          const float xs[4]  = {xv.x, xv.y, xv.z, xv.w};
          const float ws4[4] = {wv.x, wv.y, wv.z, wv.w};
#pragma unroll
          for (int c = 0; c < 4; ++c) {
            const int gi = base + li + c;
            while (gi >= nb) {
              atomicAdd(&wacc[wave][cur], acc);
              acc = 0.f;
              ++cur;
              nb = sstarts[cur + 1];
            }
            acc = fmaf(xs[c], ws4[c], acc);
          }
        }
      }
      // scalar tail (empty for the reference sizes)
      for (int li = (nIter << 7) + lane; li < cend; li += 32) {
        const int gi = base + li;
        while (gi >= nb) {
          atomicAdd(&wacc[wave][cur], acc);
          acc = 0.f;
          ++cur;
          nb = sstarts[cur + 1];
        }
        acc = fmaf(xrow[li], w_lds[li], acc);
      }
      atomicAdd(&wacc[wave][cur], acc);
    }
    __syncthreads();
    if (row < Bn) {
      float* dst = partial + ((size_t)blockIdx.x * Bn + row) * nOut;
      for (int j = lane; j < nOut; j += 32) {
        dst[j] = wacc[wave][j];
        wacc[wave][j] = 0.f;
      }
    }
    __syncthreads();
  }
}

// ---------------------------------------------------------------------------
// Kernel 2: reduce partial[nblk][E] over nblk with V_WMMA_F32_16X16X4_F32.
// A = ones(16x4)  =>  D[m,n] += B[0,n]+B[1,n]+B[2,n]+B[3,n]  (4 blocks/step).
// Sum over k is permutation-invariant, so this is robust to B row ordering.
// Each wave reduces 16 outputs; lanes 0..15 read D[0,n] from d[0].
// ---------------------------------------------------------------------------
__global__ __launch_bounds__(TPB)
void reduce_kernel(const float* __restrict__ partial, const float* __restrict__ bias,
                   float* __restrict__ out, int nblk, int E, int nOut, int nTiles) {
  const int lane = threadIdx.x & 31;
  const int wave = threadIdx.x >> 5;
  const int tile = blockIdx.x * (TPB / 32) + wave;
  if (tile >= nTiles) return;  // wave-uniform: EXEC stays all-ones for WMMA

  const int e0   = tile * 16;
  const int col  = lane & 15;
  const int half = lane >> 4;
  const int e    = e0 + col;

#if __has_builtin(__builtin_amdgcn_wmma_f32_16x16x4_f32)
  v2f a; a.x = 1.f; a.y = 1.f;                  // A = ones(16x4)
  v8f d = {0.f, 0.f, 0.f, 0.f, 0.f, 0.f, 0.f, 0.f};
  const int ng = (nblk + 3) >> 2;
  for (int g = 0; g < ng; ++g) {
    const int b0 = (g << 2) + (half << 1);
    v2f b;
    b.x = (b0 < nblk && e < E)     ? partial[(size_t)b0 * E + e]       : 0.f;
    b.y = (b0 + 1 < nblk && e < E) ? partial[(size_t)(b0 + 1) * E + e] : 0.f;
    // D = A*B + C   (8-arg form: neg_a, A, neg_b, B, c_mod, C, reuse_a, reuse_b)
    d = __builtin_amdgcn_wmma_f32_16x16x4_f32(false, a, false, b, (short)0, d,
                                              false, false);
  }
  if (lane < 16 && e < E) out[e] = d[0] + bias[e % nOut];
#else
  if (lane < 16 && e < E) {
    float s = 0.f;
    for (int b0 = 0; b0 < nblk; ++b0) s += partial[(size_t)b0 * E + e];
    out[e] = s + bias[e % nOut];
  }
#endif
}

// ---------------------------------------------------------------------------
extern "C" void kernel_launch(void* const* d_in, const int* in_sizes, int n_in,
                              void* d_out, int out_size, void* d_ws, size_t ws_size,
                              hipStream_t stream) {
  const float* x    = (const float*)d_in[0];
  const float* w    = (const float*)d_in[1];
  const float* bias = (const float*)d_in[2];
  const int*   seg  = (const int*)d_in[3];

  const int F    = in_sizes[1];             // w_flat element count
  const int Bn   = in_sizes[0] / F;         // batch
  const int nOut = in_sizes[2];             // bias element count
  float* out = (float*)d_out;

  int*   starts  = (int*)d_ws;                         // [nOut+1]
  float* partial = (float*)((char*)d_ws + 1024);       // [nblk][Bn*nOut]

  const int nblk = (F + CHUNK - 1) / CHUNK;            // 512 for reference

  seg_starts_kernel<<<512, TPB, 0, stream>>>(seg, F, starts, nOut);
  seglinear_kernel<<<nblk, TPB, 0, stream>>>(x, w, starts, partial, F, Bn, nOut);

  const int E       = Bn * nOut;
  const int nTiles  = (E + 15) / 16;
  const int blocks2 = (nTiles + (TPB / 32) - 1) / (TPB / 32);
  reduce_kernel<<<blocks2, TPB, 0, stream>>>(partial, bias, out, nblk, E, nOut,
                                             nTiles);
}